// EigenEmbedding_20109036880593
// MI455X (gfx1250) — compile-verified
//
#include <hip/hip_runtime.h>
#include <math.h>

// ---------------------------------------------------------------------------
// Batched symmetric eigendecomposition (B=512, N=256) via block Jacobi.
//   - one workgroup (256 threads = 8 wave32) per matrix
//   - A resident in LDS (256x260 f32 padded) -- exploits CDNA5's 320KB WGP LDS
//   - initial A load via gfx1250 async copy (global_load_async_to_lds_b128)
//   - 32x32 block-pair subproblems solved barrier-free inside wave 0
//     (wave32 LDS ops are in-order within a wave); waves 1-7 prefetch V
//   - rotation application (A rows, A cols, V cols) via V_WMMA_F32_16X16X4_F32
//     with all operand fetches as contiguous ds_load_b64 / global b64 and
//     accumulator writeback as ds_store_b128 (transposed staging)
//   - V maintained in d_out (L2-resident, 128MB of 192MB L2), final
//     ascending-eigenvalue column permutation staged through LDS
// ---------------------------------------------------------------------------

typedef __attribute__((ext_vector_type(2))) float v2f;
typedef __attribute__((ext_vector_type(4))) float v4f;
typedef __attribute__((ext_vector_type(8))) float v8f;

#define NN    256
#define LDA   260         // even, LDA*4 % 16 == 0 (b64/b128-aligned fragments)
#define NBLK  16          // 16 blocks of 16 per side
#define LDQ2  34          // even padded stride for 32x32 S and Q^T
#define LDR   36          // row-update staging stride  (mult of 4, !=0 mod 64)
#define LDC   260         // col/V-update staging stride (mult of 4, !=0 mod 64)
#define OUTER_SWEEPS 6
#define INNER_SWEEPS 4

// LDS layout (float offsets)
#define OFF_A    0
#define OFF_SCR  (OFF_A + NN*LDA)          // 9216-float staging (max of 256*36, 32*260)
#define OFF_S    (OFF_SCR + 9216)          // 32x34 subproblem matrix
#define OFF_QT   (OFF_S + 32*LDQ2)         // 32x34 accumulated rotation, TRANSPOSED
#define OFF_D    (OFF_QT + 32*LDQ2)        // 256 eigenvalues
#define OFF_RNK  (OFF_D + 256)             // 256 ranks (int)
#define LDS_FLOATS (OFF_RNK + 256)         // 78464 floats = 313,856 B (< 320KB)

__global__ __launch_bounds__(256)
void eigh_block_jacobi_kernel(const float* __restrict__ Ain,
                              float* __restrict__ Vout)
{
    extern __shared__ float smem[];
    float* As  = smem + OFF_A;
    float* SCR = smem + OFF_SCR;
    float* Ss  = smem + OFF_S;
    float* QTs = smem + OFF_QT;
    float* Dv  = smem + OFF_D;
    int*   Rk  = (int*)(smem + OFF_RNK);

    const int b    = blockIdx.x;
    const int tid  = threadIdx.x;
    const int wid  = tid >> 5;       // wave id 0..7
    const int lane = tid & 31;
    const int half = lane >> 4;      // lane-half selects K pair / M offset
    const int ln   = lane & 15;

    const float* Ag = Ain  + (size_t)b * (NN * NN);
    float*       Vg = Vout + (size_t)b * (NN * NN);

    // ---- async copy A -> LDS (b128/lane), overlap with V = I init ----------
    for (int idx = tid; idx < (NN * NN) / 4; idx += 256) {
        int i = idx >> 6;            // row
        int j = (idx & 63) * 4;      // col (16B aligned)
        const float* gp = Ag + i * NN + j;
        unsigned ldsA = (unsigned)(uintptr_t)(As + i * LDA + j);  // low 32b = LDS offset
        asm volatile("global_load_async_to_lds_b128 %0, %1, off"
                     :: "v"(ldsA), "v"(gp) : "memory");
        // identity init of V while the async engine streams A
        v4f iv = {0.f, 0.f, 0.f, 0.f};
        if (i >= j && i < j + 4) iv[i - j] = 1.0f;
        *(v4f*)(Vg + i * NN + j) = iv;
    }
    asm volatile("s_wait_asynccnt 0" ::: "memory");
    __threadfence();
    __syncthreads();

    auto grow = [&](int I, int J, int k) -> int {
        return (k < 16) ? (I * 16 + k) : (J * 16 + (k - 16));
    };

    for (int sweep = 0; sweep < OUTER_SWEEPS; ++sweep) {
        for (int rnd = 0; rnd < NBLK - 1; ++rnd) {
            for (int pk = 0; pk < NBLK / 2; ++pk) {
                // round-robin tournament pairing over 16 blocks
                int pI = pk, pJ = NBLK - 1 - pk;
                int I = (pI == 0) ? (NBLK - 1) : (pI - 1 + rnd) % (NBLK - 1);
                int J = (pJ - 1 + rnd) % (NBLK - 1);
                if (I > J) { int t = I; I = J; J = t; }

                // ---- 1. extract 32x32 subproblem S, QT = I -----------------
                for (int e = tid; e < 32 * 32; e += 256) {
                    int i = e >> 5, j = e & 31;
                    Ss[i * LDQ2 + j]  = As[grow(I, J, i) * LDA + grow(I, J, j)];
                    QTs[i * LDQ2 + j] = (i == j) ? 1.0f : 0.0f;
                }
                __syncthreads();

                // ---- 2. scalar Jacobi on 32x32: barrier-free in wave 0 -----
                if (wid == 0) {
                    const int pid = lane >> 1;    // 16 pairs, 2 lanes each
                    const int hh  = lane & 1;
                    for (int isw = 0; isw < INNER_SWEEPS; ++isw) {
                        for (int r = 0; r < 31; ++r) {
                            int p = (pid == 0) ? 31 : (pid - 1 + r) % 31;
                            int q = (30 - pid + r) % 31;
                            if (p > q) { int t = p; p = q; q = t; }
                            float app = Ss[p * LDQ2 + p];
                            float aqq = Ss[q * LDQ2 + q];
                            float apq = Ss[p * LDQ2 + q];
                            float c = 1.0f, s = 0.0f;
                            if (fabsf(apq) > 1e-30f) {
                                float tau = (aqq - app) / (2.0f * apq);
                                float tt  = copysignf(1.0f, tau) /
                                            (fabsf(tau) + sqrtf(1.0f + tau * tau));
                                c = 1.0f / sqrtf(1.0f + tt * tt);
                                s = tt * c;
                            }
                            // column rotation on S, row rotation on QT
                            // (wave-internal LDS ordering: program order)
                            for (int l = hh * 16; l < hh * 16 + 16; ++l) {
                                float sp = Ss[l * LDQ2 + p], sq = Ss[l * LDQ2 + q];
                                Ss[l * LDQ2 + p] = c * sp - s * sq;
                                Ss[l * LDQ2 + q] = s * sp + c * sq;
                                float qp = QTs[p * LDQ2 + l], qq = QTs[q * LDQ2 + l];
                                QTs[p * LDQ2 + l] = c * qp - s * qq;
                                QTs[q * LDQ2 + l] = s * qp + c * qq;
                            }
                            // row rotation on S
                            for (int cc = hh * 16; cc < hh * 16 + 16; ++cc) {
                                float sp = Ss[p * LDQ2 + cc], sq = Ss[q * LDQ2 + cc];
                                Ss[p * LDQ2 + cc] = c * sp - s * sq;
                                Ss[q * LDQ2 + cc] = s * sp + c * sq;
                            }
                        }
                    }
                } else {
                    // waves 1-7: prefetch the V rows the V-GEMM will read
                    for (int row = tid - 32; row < NN; row += 224) {
                        __builtin_prefetch(&Vg[row * NN + I * 16], 0, 0);
                        __builtin_prefetch(&Vg[row * NN + J * 16], 0, 0);
                    }
                }
                __syncthreads();

                // ---- 3. row update: R = Q^T * A([I,J], :)  (32 x 256) ------
                // B-fragment read via symmetry: A(g,c) == A(c,g) here.
                for (int t = wid; t < 32; t += 8) {
                    int rt = t & 1, ct = t >> 1;
                    v8f acc = {};
#pragma unroll
                    for (int kc = 0; kc < 8; ++kc) {
                        int ka = kc * 4 + 2 * half;                  // even
                        int g0 = grow(I, J, ka);                     // even, g1=g0+1
                        v2f a  = *(const v2f*)&QTs[(rt * 16 + ln) * LDQ2 + ka];
                        v2f bb = *(const v2f*)&As[(ct * 16 + ln) * LDA + g0];
                        acc = __builtin_amdgcn_wmma_f32_16x16x4_f32(
                                  false, a, false, bb, (short)0, acc, false, false);
                    }
                    int base = (ct * 16 + ln) * LDR + rt * 16 + half * 8;
                    *(v4f*)&SCR[base]     = (v4f){acc[0], acc[1], acc[2], acc[3]};
                    *(v4f*)&SCR[base + 4] = (v4f){acc[4], acc[5], acc[6], acc[7]};
                }
                __syncthreads();
                for (int e = tid; e < 32 * 256; e += 256) {
                    int r = e >> 8, cc = e & 255;
                    As[grow(I, J, r) * LDA + cc] = SCR[cc * LDR + r];
                }
                __syncthreads();

                // ---- 4. column update: C = A(:, [I,J]) * Q  (256 x 32) -----
                for (int t = wid; t < 32; t += 8) {
                    int rt = t >> 1, ct = t & 1;
                    v8f acc = {};
#pragma unroll
                    for (int kc = 0; kc < 8; ++kc) {
                        int ka = kc * 4 + 2 * half;
                        int g0 = grow(I, J, ka);
                        v2f a  = *(const v2f*)&As[(rt * 16 + ln) * LDA + g0];
                        v2f bb = *(const v2f*)&QTs[(ct * 16 + ln) * LDQ2 + ka];
                        acc = __builtin_amdgcn_wmma_f32_16x16x4_f32(
                                  false, a, false, bb, (short)0, acc, false, false);
                    }
                    int base = (ct * 16 + ln) * LDC + rt * 16 + half * 8;
                    *(v4f*)&SCR[base]     = (v4f){acc[0], acc[1], acc[2], acc[3]};
                    *(v4f*)&SCR[base + 4] = (v4f){acc[4], acc[5], acc[6], acc[7]};
                }
                __syncthreads();
                for (int e = tid; e < 256 * 32; e += 256) {
                    int r = e & 255, cc = e >> 8;
                    As[r * LDA + grow(I, J, cc)] = SCR[cc * LDC + r];
                }
                __syncthreads();

                // ---- 5. eigenvector update: V(:, [I,J]) *= Q ---------------
                for (int t = wid; t < 32; t += 8) {
                    int rt = t >> 1, ct = t & 1;
                    v8f acc = {};
#pragma unroll
                    for (int kc = 0; kc < 8; ++kc) {
                        int ka = kc * 4 + 2 * half;
                        int g0 = grow(I, J, ka);
                        v2f a  = *(const v2f*)&Vg[(rt * 16 + ln) * NN + g0];
                        v2f bb = *(const v2f*)&QTs[(ct * 16 + ln) * LDQ2 + ka];
                        acc = __builtin_amdgcn_wmma_f32_16x16x4_f32(
                                  false, a, false, bb, (short)0, acc, false, false);
                    }
                    int base = (ct * 16 + ln) * LDC + rt * 16 + half * 8;
                    *(v4f*)&SCR[base]     = (v4f){acc[0], acc[1], acc[2], acc[3]};
                    *(v4f*)&SCR[base + 4] = (v4f){acc[4], acc[5], acc[6], acc[7]};
                }
                __syncthreads();
                for (int e = tid; e < 256 * 32; e += 256) {
                    int cc = e & 31, r = e >> 5;
                    Vg[r * NN + grow(I, J, cc)] = SCR[cc * LDC + r];
                }
                __threadfence();
                __syncthreads();
            }
        }
    }

    // ---- final: ascending-eigenvalue rank sort + column permutation --------
    Dv[tid] = As[tid * LDA + tid];
    __syncthreads();
    {
        float dj = Dv[tid];
        int r = 0;
        for (int k = 0; k < NN; ++k) {
            float dk = Dv[k];
            r += (dk < dj) || (dk == dj && k < tid);
        }
        Rk[tid] = r;
    }
    __syncthreads();
    // stage V through the (now free) LDS A buffer, then scatter columns
    for (int idx = tid; idx < (NN * NN) / 4; idx += 256) {
        int i = idx >> 6, j = (idx & 63) * 4;
        *(v4f*)&As[i * LDA + j] = *(const v4f*)&Vg[i * NN + j];
    }
    __syncthreads();
    for (int idx = tid; idx < NN * NN; idx += 256) {
        int i = idx >> 8, j = idx & 255;
        Vg[i * NN + Rk[j]] = As[i * LDA + j];
    }
}

extern "C" void kernel_launch(void* const* d_in, const int* in_sizes, int n_in,
                              void* d_out, int out_size, void* d_ws, size_t ws_size,
                              hipStream_t stream) {
    (void)d_ws; (void)ws_size; (void)n_in; (void)out_size;
    const float* A = (const float*)d_in[0];
    float* V = (float*)d_out;
    int B = in_sizes[0] / (NN * NN);           // 512
    size_t lds_bytes = (size_t)LDS_FLOATS * sizeof(float);  // ~306 KB of 320 KB WGP LDS
    eigh_block_jacobi_kernel<<<B, 256, lds_bytes, stream>>>(A, V);
}